// ConvNextMaskRCNNSingleRoIExtractor_8873402434045
// MI455X (gfx1250) — compile-verified
//
#include <hip/hip_runtime.h>
#include <math.h>

// FPN RoIAlign (SingleRoIExtractor): per-ROI level dispatch + RoIAlign 7x7, SR=2,
// aligned=True, f32. Pooling (mean over 2x2 samples) done as an exact f32 matmul
// on the CDNA5 matrix cores: D[49,C] = A[49,196] x S[196,C] via V_WMMA_F32_16X16X4_F32.
//
// Phase-2: wave-uniform (readfirstlane) tile/trip scalars -> scalar loop control,
// contiguous k-range per m-tile (no acc phi copies), A fragment = two integer
// compares, unconditional b128-mergeable stores for m-tiles 0..2.

#define ROUT 7
#define NSX 14            // ROUT * SR samples per axis
#define NSAMP 196         // NSX * NSX  (== 4 * 49, so the K dim is 49 steps of 4)
#define C_TOTAL 256
#define C_CHUNK 64        // channels staged in LDS per pass
#define SSTRIDE 73        // LDS row stride (floats): odd (conflict-free stores),
                          // 2*73 mod 64 = 18 -> lane halves hit disjoint banks on B loads
#define NTHREADS 256

typedef __attribute__((ext_vector_type(2))) float v2f;
typedef __attribute__((ext_vector_type(8))) float v8f;

__global__ __launch_bounds__(NTHREADS)
void roi_align_fpn_wmma(const float* __restrict__ f0,
                        const float* __restrict__ f1,
                        const float* __restrict__ f2,
                        const float* __restrict__ f3,
                        const float* __restrict__ rois,
                        float* __restrict__ out,
                        int K, int H0, int H1, int H2, int H3)
{
    __shared__ float S[NSAMP * SSTRIDE];          // sample matrix S[s][c_local]
    __shared__ int   xlo[NSX], xhi[NSX], ylo[NSX], yhi[NSX];
    __shared__ float xw0[NSX], xw1[NSX], yw0[NSX], yw1[NSX];

    const int roi = blockIdx.x;
    if (roi >= K) return;
    const int tid = threadIdx.x;

    // ---- per-ROI uniform setup ----
    const float r0 = rois[roi * 5 + 0];
    const float x1 = rois[roi * 5 + 1];
    const float y1 = rois[roi * 5 + 2];
    const float x2 = rois[roi * 5 + 3];
    const float y2 = rois[roi * 5 + 4];
    const int b = (int)r0;

    const float scale = sqrtf((x2 - x1) * (y2 - y1));
    int lvl = (int)floorf(log2f(scale * (1.0f / 56.0f) + 1e-6f));
    lvl = min(max(lvl, 0), 3);

    const float* feat; int H;
    switch (lvl) {
        case 0:  feat = f0; H = H0; break;
        case 1:  feat = f1; H = H1; break;
        case 2:  feat = f2; H = H2; break;
        default: feat = f3; H = H3; break;
    }
    const int W = H;
    const float ss = 1.0f / (float)(4 << lvl);
    const float sx = x1 * ss - 0.5f;
    const float sy = y1 * ss - 0.5f;
    const float bw = (x2 - x1) * ss * (1.0f / ROUT);
    const float bh = (y2 - y1) * ss * (1.0f / ROUT);
    const int imgBase = b * C_TOTAL * H * W;

    // ---- per-axis bilinear tables (torchvision _axis_coords; validity folded
    //      into the weights so invalid samples become exact zeros) ----
    if (tid < 2 * NSX) {
        const bool isY = tid >= NSX;
        const int i = isY ? tid - NSX : tid;
        const int size = isY ? H : W;
        const float start = isY ? sy : sx;
        const float bwd = isY ? bh : bw;
        const int bin = i >> 1, j = i & 1;
        const float c = start + bwd * ((float)bin + 0.25f + 0.5f * (float)j);
        const float valid = (c >= -1.0f && c <= (float)size) ? 1.0f : 0.0f;
        float cc = fminf(fmaxf(c, 0.0f), (float)(size - 1));
        const int lo0 = (int)floorf(cc);
        const bool cap = lo0 >= size - 1;
        const int lo = cap ? size - 1 : lo0;
        const int hi = cap ? size - 1 : lo0 + 1;
        const float fr = cap ? 0.0f : (cc - (float)lo0);
        const float w0 = valid * (1.0f - fr);
        const float w1 = valid * fr;
        if (isY) { ylo[i] = lo; yhi[i] = hi; yw0[i] = w0; yw1[i] = w1; }
        else     { xlo[i] = lo; xhi[i] = hi; xw0[i] = w0; xw1[i] = w1; }
    }

    const int wv   = tid >> 5;          // wave id (wave32)
    const int lane = tid & 31;
    const int half = lane >> 4;         // 0: lanes 0-15, 1: lanes 16-31
    const int nloc = lane & 15;

    for (int chunk = 0; chunk < C_TOTAL / C_CHUNK; ++chunk) {
        __syncthreads();  // axis tables ready / previous chunk's S fully consumed

        // ---- Phase 1: gather+bilinear -> S[s][c_local] (lanes stride along s
        //      so the 4 taps are row-coalesced in global memory) ----
        for (int idx = tid; idx < NSAMP * C_CHUNK; idx += NTHREADS) {
            const int cl = idx / NSAMP;
            const int s  = idx - cl * NSAMP;
            const int syi = s / NSX;
            const int sxi = s - syi * NSX;
            const int cbase = imgBase + (chunk * C_CHUNK + cl) * H * W;
            const float* p0 = feat + cbase + ylo[syi] * W;
            const float* p1 = feat + cbase + yhi[syi] * W;
            const int cx0 = xlo[sxi], cx1 = xhi[sxi];
            const float va = p0[cx0], vb = p0[cx1];
            const float vc = p1[cx0], vd = p1[cx1];
            const float hx0 = xw0[sxi], hx1 = xw1[sxi];
            S[s * SSTRIDE + cl] =
                yw0[syi] * (hx0 * va + hx1 * vb) +
                yw1[syi] * (hx0 * vc + hx1 * vd);
        }
        __syncthreads();

        // ---- Phase 2: pooling matmul D = A x S via f32 WMMA 16x16x4 ----
        // Tiles: 4 m-tiles (49 bins padded to 64) x 4 n-tiles (64 channels).
        for (int t = wv; t < 16; t += NTHREADS / 32) {
            // t is wave-uniform; force it scalar so tile params & loop bounds
            // live in SGPRs (scalar loop control, scalar LDS base stepping).
            const int tuni  = __builtin_amdgcn_readfirstlane(t);
            const int mtile = tuni >> 2;
            const int ntile = tuni & 3;
            const int binlo_t = mtile * 16;

            // Contiguous k-step range touched by this m-tile:
            // bins [b0, min(b0+15,48)] -> ph in [b0/7, b1/7] -> kk in [7*phlo, 7*phhi+6]
            const int kks = 7 * (binlo_t / ROUT);
            const int kke = 7 * (min(binlo_t + 15, 48) / ROUT) + 6;  // trip = 21 or 7

            // A-fragment lane constants: M = nloc (ISA 16x4 f32 layout).
            // Row 'mybin' of A is nonzero at samples {s00, s00+1, s00+14, s00+15}.
            const int mybin = binlo_t + nloc;
            const bool mvalid = mybin < 49;
            const int ph = mybin / ROUT;
            const int pw = mybin - ph * ROUT;
            const int s00 = mvalid ? (28 * ph + 2 * pw) : -2;  // -2: never matches
            const int s10 = s00 + NSX;

            const int n = ntile * 16 + nloc;
            const float* sp = &S[(4 * kks + 2 * half) * SSTRIDE + n];
            int t0 = 4 * kks + 2 * half;   // even sample index of this lane's slot pair

            v8f acc = {0.f, 0.f, 0.f, 0.f, 0.f, 0.f, 0.f, 0.f};

            #pragma unroll 7
            for (int kk = kks; kk <= kke; ++kk) {
                // A fragment: af.x == af.y (t0 even, t0+1 odd hit the same pair)
                const float a = (t0 == s00 || t0 == s10) ? 0.25f : 0.0f;
                v2f af; af.x = a; af.y = a;

                v2f bf;
                bf.x = sp[0];
                bf.y = sp[SSTRIDE];

                acc = __builtin_amdgcn_wmma_f32_16x16x4_f32(
                    false, af, false, bf, (short)0, acc, false, false);

                sp += 4 * SSTRIDE;
                t0 += 4;
            }

            // D layout: element r -> M = r + 8*half, N = nloc.
            // Bins for this lane are binlo_t + 8*half + (0..7): 32 consecutive bytes.
            const int cg = chunk * C_CHUNK + ntile * 16 + nloc;
            float* op = out + roi * (C_TOTAL * ROUT * ROUT) + cg * (ROUT * ROUT)
                            + binlo_t + 8 * half;
            if (binlo_t < 48) {
                // m-tiles 0..2: every bin < 49 -> unconditional, b128-mergeable
                #pragma unroll
                for (int r = 0; r < 8; ++r) op[r] = acc[r];
            } else if (half == 0) {
                // m-tile 3: only bin 48 is real (r==0, lanes 0-15)
                op[0] = acc[0];
            }
        }
    }
}

extern "C" void kernel_launch(void* const* d_in, const int* in_sizes, int n_in,
                              void* d_out, int out_size, void* d_ws, size_t ws_size,
                              hipStream_t stream) {
    const float* f0 = (const float*)d_in[0];
    const float* f1 = (const float*)d_in[1];
    const float* f2 = (const float*)d_in[2];
    const float* f3 = (const float*)d_in[3];
    const float* rois = (const float*)d_in[4];
    float* out = (float*)d_out;

    // feat_i is [2, 256, H, H] -> H = sqrt(elems / 512)
    auto hw = [](int elems) {
        int h = (int)(sqrtf((float)(elems / 512)) + 0.5f);
        return h;
    };
    const int H0 = hw(in_sizes[0]);
    const int H1 = hw(in_sizes[1]);
    const int H2 = hw(in_sizes[2]);
    const int H3 = hw(in_sizes[3]);
    const int K = in_sizes[4] / 5;

    hipLaunchKernelGGL(roi_align_fpn_wmma, dim3(K), dim3(NTHREADS), 0, stream,
                       f0, f1, f2, f3, rois, out, K, H0, H1, H2, H3);
}